// NTM_50672024158923
// MI455X (gfx1250) — compile-verified
//
#include <hip/hip_runtime.h>
#include <math.h>

// Problem sizes (compile-time constants from the reference)
#define B_ 256
#define D_ 256
#define H_ 512
#define M_ 64
#define N_ 2048

typedef __attribute__((ext_vector_type(2))) float v2f;
typedef __attribute__((ext_vector_type(8))) float v8f;

__device__ __forceinline__ float sigmoidf_(float x) { return 1.0f / (1.0f + expf(-x)); }

// ---------------------------------------------------------------------------
// WMMA f32 GEMM:  Cout[M x N] = A[M x K] * W[N x K]^T (+ bias[N]) (+ Cin) ; act
// Each wave computes a 16x64 strip: one A fragment reused across four B
// fragments -> 4x V_WMMA_F32_16X16X4_F32 per A load. Fragments for step k+4
// are loaded BEFORE the step-k WMMAs (explicit double buffering) so the
// scheduler can overlap global loads with the XDL pipe instead of emitting
// s_wait_loadcnt 0 before every WMMA.
// A-matrix lane layout (ISA 7.12.2, 32-bit A 16x4): lane l holds row m=l&15,
//   VGPR0 = K = 2*(l>>4), VGPR1 = K = 2*(l>>4)+1.  B (4x16) mirrors this;
//   since B[k][n] = W[n][k], W addressing is identical to A's.
// C/D layout: VGPR r -> row = r + 8*(l>>4), col = l&15.
// colGroups = (N/16)/4 ; waves = (M/16)*colGroups ; grid.x = waves/8.
// ---------------------------------------------------------------------------
__global__ void gemm_wmma_kernel(const float* __restrict__ A, int lda,
                                 const float* __restrict__ W, int ldw,
                                 const float* __restrict__ bias,
                                 const float* __restrict__ Cin,
                                 float* __restrict__ Cout, int ldc,
                                 int colGroups, int K, int act)
{
    const int wave = threadIdx.x >> 5;
    const int lane = threadIdx.x & 31;
    const int task = blockIdx.x * 8 + wave;
    const int tNg = task % colGroups;   // 64-wide column group
    const int tM  = task / colGroups;   // 16-row tile

    const int r0 = lane & 15;
    const int kh = (lane >> 4) * 2;
    const float* Arow  = A + (size_t)(tM * 16 + r0) * lda + kh;
    const float* Wrow0 = W + (size_t)(tNg * 64 + r0) * ldw + kh;
    const float* Wrow1 = Wrow0 + (size_t)16 * ldw;
    const float* Wrow2 = Wrow0 + (size_t)32 * ldw;
    const float* Wrow3 = Wrow0 + (size_t)48 * ldw;

    v8f acc0 = {0.f,0.f,0.f,0.f,0.f,0.f,0.f,0.f};
    v8f acc1 = acc0, acc2 = acc0, acc3 = acc0;

    // prologue: fragments for k = 0
    v2f a, b0, b1, b2, b3;
    a.x  = Arow[0];   a.y  = Arow[1];
    b0.x = Wrow0[0];  b0.y = Wrow0[1];
    b1.x = Wrow1[0];  b1.y = Wrow1[1];
    b2.x = Wrow2[0];  b2.y = Wrow2[1];
    b3.x = Wrow3[0];  b3.y = Wrow3[1];

    for (int k = 0; k + 4 < K; k += 4) {
        // issue next iteration's loads first (independent of current WMMAs)
        const int kn = k + 4;
        v2f an, bn0, bn1, bn2, bn3;
        an.x  = Arow[kn];   an.y  = Arow[kn + 1];
        bn0.x = Wrow0[kn];  bn0.y = Wrow0[kn + 1];
        bn1.x = Wrow1[kn];  bn1.y = Wrow1[kn + 1];
        bn2.x = Wrow2[kn];  bn2.y = Wrow2[kn + 1];
        bn3.x = Wrow3[kn];  bn3.y = Wrow3[kn + 1];

        acc0 = __builtin_amdgcn_wmma_f32_16x16x4_f32(false, a, false, b0, (short)0, acc0, false, false);
        acc1 = __builtin_amdgcn_wmma_f32_16x16x4_f32(false, a, false, b1, (short)0, acc1, false, false);
        acc2 = __builtin_amdgcn_wmma_f32_16x16x4_f32(false, a, false, b2, (short)0, acc2, false, false);
        acc3 = __builtin_amdgcn_wmma_f32_16x16x4_f32(false, a, false, b3, (short)0, acc3, false, false);

        a = an; b0 = bn0; b1 = bn1; b2 = bn2; b3 = bn3;
    }
    // epilogue: last K-step
    acc0 = __builtin_amdgcn_wmma_f32_16x16x4_f32(false, a, false, b0, (short)0, acc0, false, false);
    acc1 = __builtin_amdgcn_wmma_f32_16x16x4_f32(false, a, false, b1, (short)0, acc1, false, false);
    acc2 = __builtin_amdgcn_wmma_f32_16x16x4_f32(false, a, false, b2, (short)0, acc2, false, false);
    acc3 = __builtin_amdgcn_wmma_f32_16x16x4_f32(false, a, false, b3, (short)0, acc3, false, false);

    const int rowBase = tM * 16 + 8 * (lane >> 4);
#pragma unroll
    for (int t = 0; t < 4; ++t) {
        const v8f acc = (t == 0) ? acc0 : (t == 1) ? acc1 : (t == 2) ? acc2 : acc3;
        const int col = tNg * 64 + t * 16 + (lane & 15);
        const float bv = bias ? bias[col] : 0.0f;
#pragma unroll
        for (int r = 0; r < 8; ++r) {
            const int row = rowBase + r;
            const size_t idx = (size_t)row * ldc + col;
            float v = acc[r] + bv;
            if (Cin) v += Cin[idx];
            if (act == 1) v = tanhf(v);
            Cout[idx] = v;
        }
    }
}

// ---------------------------------------------------------------------------
// Head scalar params: c = h@Wc^T + bc  -> beta,g,gamma ;  s = softmax(h@Ws^T+bs)
// One wave per batch row (outputs are only 6 wide -> not worth WMMA).
// ---------------------------------------------------------------------------
__global__ void head_scalars_kernel(const float* __restrict__ h,
                                    const float* __restrict__ Wc, const float* __restrict__ bc,
                                    const float* __restrict__ Ws, const float* __restrict__ bs,
                                    float* __restrict__ beta, float* __restrict__ g,
                                    float* __restrict__ gamma, float* __restrict__ s)
{
    const int b = blockIdx.x, lane = threadIdx.x;
    float acc[6] = {0.f, 0.f, 0.f, 0.f, 0.f, 0.f};
    const float* hb = h + (size_t)b * H_;
    for (int k = lane; k < H_; k += 32) {
        const float hv = hb[k];
#pragma unroll
        for (int j = 0; j < 3; ++j) {
            acc[j]     += hv * Wc[j * H_ + k];
            acc[3 + j] += hv * Ws[j * H_ + k];
        }
    }
#pragma unroll
    for (int j = 0; j < 6; ++j)
        for (int off = 16; off; off >>= 1) acc[j] += __shfl_xor(acc[j], off, 32);
    if (lane == 0) {
        const float c0 = acc[0] + bc[0], c1 = acc[1] + bc[1], c2 = acc[2] + bc[2];
        beta[b]  = fmaxf(c0, 0.0f) + 1e-4f;
        g[b]     = sigmoidf_(c1);
        gamma[b] = fmaxf(c2, 0.0f) + 1.0001f;
        const float l0 = acc[3] + bs[0], l1 = acc[4] + bs[1], l2 = acc[5] + bs[2];
        const float mx = fmaxf(l0, fmaxf(l1, l2));
        const float e0 = expf(l0 - mx), e1 = expf(l1 - mx), e2 = expf(l2 - mx);
        const float inv = 1.0f / (e0 + e1 + e2);
        s[b * 3 + 0] = e0 * inv;
        s[b * 3 + 1] = e1 * inv;
        s[b * 3 + 2] = e2 * inv;
    }
}

// ---------------------------------------------------------------------------
// Fused NTM addressing for one batch row per block (256 thr = 8 waves):
//   cos sim -> softmax(beta*cos) -> interpolate -> circular shift -> ^gamma
//   -> normalize ; optionally M_read = head @ memory.
// memory rows (64 f32 = 256B) are read wave-cooperatively (lane -> float2).
// ---------------------------------------------------------------------------
__global__ void address_kernel(const float* __restrict__ memory,
                               const float* __restrict__ prev_head,
                               const float* __restrict__ kk,
                               const float* __restrict__ beta_, const float* __restrict__ g_,
                               const float* __restrict__ gamma_, const float* __restrict__ s_,
                               float* __restrict__ head_out,
                               float* __restrict__ m_read /* may be null */)
{
    __shared__ float kks[M_];
    __shared__ float buf[N_];
    __shared__ float inter[N_];
    __shared__ float red[8];
    __shared__ float mp[8][M_];

    const int b = blockIdx.x;
    const int tid = threadIdx.x, wave = tid >> 5, lane = tid & 31;

    if (tid < M_) kks[tid] = kk[(size_t)b * M_ + tid];
    __syncthreads();

    float nk2 = 0.f;
#pragma unroll
    for (int m = 0; m < M_; ++m) nk2 += kks[m] * kks[m];
    const float nk = sqrtf(nk2);

    const float beta = beta_[b], gg = g_[b], gamma = gamma_[b];
    const float s0 = s_[b * 3], s1 = s_[b * 3 + 1], s2 = s_[b * 3 + 2];
    const float* memb = memory + (size_t)b * N_ * M_;

    // cosine similarity logits
    for (int n = wave; n < N_; n += 8) {
        const float2* row = (const float2*)(memb + (size_t)n * M_);
        __builtin_prefetch(memb + (size_t)(n + 8) * M_, 0, 1);
        const float2 v = row[lane];
        float dot = v.x * kks[lane * 2] + v.y * kks[lane * 2 + 1];
        float ss  = v.x * v.x + v.y * v.y;
        for (int off = 16; off; off >>= 1) {
            dot += __shfl_xor(dot, off, 32);
            ss  += __shfl_xor(ss,  off, 32);
        }
        if (lane == 0) {
            const float nM = sqrtf(ss);
            buf[n] = beta * (dot / fmaxf(nk * nM, 1e-8f));
        }
    }
    __syncthreads();

    // softmax over N
    float lmax = -3.4e38f;
    for (int i = tid; i < N_; i += 256) lmax = fmaxf(lmax, buf[i]);
    for (int off = 16; off; off >>= 1) lmax = fmaxf(lmax, __shfl_xor(lmax, off, 32));
    if (lane == 0) red[wave] = lmax;
    __syncthreads();
    float gmax = red[0];
#pragma unroll
    for (int w = 1; w < 8; ++w) gmax = fmaxf(gmax, red[w]);
    __syncthreads();

    float lsum = 0.f;
    for (int i = tid; i < N_; i += 256) { const float e = expf(buf[i] - gmax); buf[i] = e; lsum += e; }
    for (int off = 16; off; off >>= 1) lsum += __shfl_xor(lsum, off, 32);
    if (lane == 0) red[wave] = lsum;
    __syncthreads();
    float gsum = 0.f;
#pragma unroll
    for (int w = 0; w < 8; ++w) gsum += red[w];
    const float inv = 1.0f / gsum;
    __syncthreads();

    // interpolate with previous head
    const float* phb = prev_head + (size_t)b * N_;
    for (int i = tid; i < N_; i += 256)
        inter[i] = gg * (buf[i] * inv) + (1.0f - gg) * phb[i];
    __syncthreads();

    // circular shift (SHIFTS = 1,0,-1) + sharpen
    float lsum2 = 0.f;
    for (int i = tid; i < N_; i += 256) {
        float o = s0 * inter[(i + 1) & (N_ - 1)] + s1 * inter[i] + s2 * inter[(i - 1) & (N_ - 1)];
        o = powf(o, gamma);
        buf[i] = o;
        lsum2 += o;
    }
    for (int off = 16; off; off >>= 1) lsum2 += __shfl_xor(lsum2, off, 32);
    if (lane == 0) red[wave] = lsum2;
    __syncthreads();
    float gsum2 = 0.f;
#pragma unroll
    for (int w = 0; w < 8; ++w) gsum2 += red[w];
    const float inv2 = 1.0f / gsum2;

    for (int i = tid; i < N_; i += 256) {
        const float hv = buf[i] * inv2;
        buf[i] = hv;
        head_out[(size_t)b * N_ + i] = hv;
    }
    __syncthreads();

    // optional M_read = head @ memory  (memory rows are L2-hot from pass 1)
    if (m_read) {
        float ax = 0.f, ay = 0.f;
        for (int n = wave; n < N_; n += 8) {
            const float2* row = (const float2*)(memb + (size_t)n * M_);
            const float2 v = row[lane];
            const float w = buf[n];
            ax += w * v.x; ay += w * v.y;
        }
        mp[wave][lane * 2]     = ax;
        mp[wave][lane * 2 + 1] = ay;
        __syncthreads();
        if (tid < M_) {
            float sacc = 0.f;
#pragma unroll
            for (int w = 0; w < 8; ++w) sacc += mp[w][tid];
            m_read[(size_t)b * M_ + tid] = sacc;
        }
    }
}

// ---------------------------------------------------------------------------
// LSTM pointwise gates
// ---------------------------------------------------------------------------
__global__ void lstm_kernel(const float* __restrict__ z, const float* __restrict__ c_tm1,
                            float* __restrict__ h_out, float* __restrict__ c_out)
{
    const int i = blockIdx.x * blockDim.x + threadIdx.x;  // over B*H
    const int b = i / H_, j = i % H_;
    const float* zb = z + (size_t)b * 4 * H_;
    const float zi = zb[j], zf = zb[H_ + j], zg = zb[2 * H_ + j], zo = zb[3 * H_ + j];
    const float c = sigmoidf_(zf) * c_tm1[i] + sigmoidf_(zi) * tanhf(zg);
    const float h = sigmoidf_(zo) * tanhf(c);
    c_out[i] = c;
    h_out[i] = h;
}

// ---------------------------------------------------------------------------
// e = sigmoid(wa[:, :64]),  a = tanh(wa[:, 64:])
// ---------------------------------------------------------------------------
__global__ void erase_add_kernel(const float* __restrict__ wa,
                                 float* __restrict__ e, float* __restrict__ a)
{
    const int i = blockIdx.x * blockDim.x + threadIdx.x;  // over B*128
    const int b = i >> 7, j = i & 127;
    const float v = wa[i];
    if (j < 64) e[b * 64 + j]        = sigmoidf_(v);
    else        a[b * 64 + (j - 64)] = tanhf(v);
}

// ---------------------------------------------------------------------------
// M_out = memory * (1 - ww*e) + ww*a  (float4-vectorized, 134 MB store)
// ---------------------------------------------------------------------------
__global__ void mem_update_kernel(const float* __restrict__ memory,
                                  const float* __restrict__ whead,
                                  const float* __restrict__ e, const float* __restrict__ a,
                                  float* __restrict__ m_out)
{
    const size_t i4 = (size_t)blockIdx.x * blockDim.x + threadIdx.x;  // over B*N*16
    const int m4 = (int)(i4 & 15);
    const size_t bn = i4 >> 4;
    const int b = (int)(bn >> 11);  // N = 2048 = 2^11
    const float ww = whead[bn];
    const float4 mv = ((const float4*)memory)[i4];
    const float4 ev = ((const float4*)(e + (size_t)b * 64))[m4];
    const float4 av = ((const float4*)(a + (size_t)b * 64))[m4];
    float4 o;
    o.x = mv.x * (1.0f - ww * ev.x) + ww * av.x;
    o.y = mv.y * (1.0f - ww * ev.y) + ww * av.y;
    o.z = mv.z * (1.0f - ww * ev.z) + ww * av.z;
    o.w = mv.w * (1.0f - ww * ev.w) + ww * av.w;
    ((float4*)m_out)[i4] = o;
}

// ---------------------------------------------------------------------------
extern "C" void kernel_launch(void* const* d_in, const int* in_sizes, int n_in,
                              void* d_out, int out_size, void* d_ws, size_t ws_size,
                              hipStream_t stream)
{
    const float* inp    = (const float*)d_in[0];
    const float* h_tm1  = (const float*)d_in[1];
    const float* c_tm1  = (const float*)d_in[2];
    const float* rh_tm1 = (const float*)d_in[3];
    const float* wh_tm1 = (const float*)d_in[4];
    const float* memory = (const float*)d_in[5];
    const float* W_ih   = (const float*)d_in[6];
    const float* W_hh   = (const float*)d_in[7];
    const float* b_ih   = (const float*)d_in[8];
    const float* b_hh   = (const float*)d_in[9];
    const float* W_rk   = (const float*)d_in[10];
    const float* b_rk   = (const float*)d_in[11];
    const float* W_wk   = (const float*)d_in[12];
    const float* b_wk   = (const float*)d_in[13];
    const float* W_rc   = (const float*)d_in[14];
    const float* b_rc   = (const float*)d_in[15];
    const float* W_wc   = (const float*)d_in[16];
    const float* b_wc   = (const float*)d_in[17];
    const float* W_rs   = (const float*)d_in[18];
    const float* b_rs   = (const float*)d_in[19];
    const float* W_ws   = (const float*)d_in[20];
    const float* b_ws   = (const float*)d_in[21];
    const float* W_w    = (const float*)d_in[22];
    const float* b_w    = (const float*)d_in[23];

    float* out   = (float*)d_out;                 // return order, flat:
    float* h_t   = out;                           // (B,H)   = 131072
    float* c_t   = out + 131072;                  // (B,H)   = 131072
    float* rhead = out + 262144;                  // (B,N)   = 524288
    float* whead = out + 786432;                  // (B,N)   = 524288
    float* m_out = out + 1310720;                 // (B,N,M) = 33554432

    float* ws      = (float*)d_ws;
    float* kk_r    = ws;            // 16384
    float* beta_r  = ws + 16384;    // 256
    float* g_r     = ws + 16640;    // 256
    float* gamma_r = ws + 16896;    // 256
    float* s_r     = ws + 17152;    // 768
    float* kk_w    = ws + 17920;    // 16384
    float* beta_w  = ws + 34304;    // 256
    float* g_w     = ws + 34560;    // 256
    float* gamma_w = ws + 34816;    // 256
    float* s_w     = ws + 35072;    // 768
    float* m_read  = ws + 35840;    // 16384
    float* z       = ws + 52224;    // 524288
    float* wa      = ws + 576512;   // 32768
    float* e       = ws + 609280;   // 16384
    float* a       = ws + 625664;   // 16384  (total ~2.57 MB)

    // Wave-task counts: (rows/16) * (cols/64); grid.x = tasks/8 (8 waves/block)
    // kk (256x64):  16*1  = 16 tasks  -> 2 blocks
    // z  (256x2048):16*32 = 512 tasks -> 64 blocks
    // wa (256x128): 16*2  = 32 tasks  -> 4 blocks

    // ---- read head ----
    head_scalars_kernel<<<B_, 32, 0, stream>>>(h_tm1, W_rc, b_rc, W_rs, b_rs,
                                               beta_r, g_r, gamma_r, s_r);
    gemm_wmma_kernel<<<2, 256, 0, stream>>>(h_tm1, H_, W_rk, H_, b_rk, nullptr,
                                            kk_r, M_, /*colGroups=*/1, H_, /*act=*/1);
    address_kernel<<<B_, 256, 0, stream>>>(memory, rh_tm1, kk_r,
                                           beta_r, g_r, gamma_r, s_r, rhead, m_read);

    // ---- LSTM pre-activations: z = inp@Wih[:, :D]^T + M_read@Wih[:, D:]^T + h@Whh^T + b ----
    gemm_wmma_kernel<<<64, 256, 0, stream>>>(inp,    D_, W_ih,       320, b_ih, nullptr,
                                             z, 4 * H_, 32, D_, 0);
    gemm_wmma_kernel<<<64, 256, 0, stream>>>(m_read, M_, W_ih + 256, 320, nullptr, z,
                                             z, 4 * H_, 32, M_, 0);
    gemm_wmma_kernel<<<64, 256, 0, stream>>>(h_tm1,  H_, W_hh,       H_,  b_hh, z,
                                             z, 4 * H_, 32, H_, 0);
    lstm_kernel<<<(B_ * H_) / 256, 256, 0, stream>>>(z, c_tm1, h_t, c_t);

    // ---- write head (from h_t) ----
    head_scalars_kernel<<<B_, 32, 0, stream>>>(h_t, W_wc, b_wc, W_ws, b_ws,
                                               beta_w, g_w, gamma_w, s_w);
    gemm_wmma_kernel<<<2, 256, 0, stream>>>(h_t, H_, W_wk, H_, b_wk, nullptr,
                                            kk_w, M_, 1, H_, 1);
    address_kernel<<<B_, 256, 0, stream>>>(memory, wh_tm1, kk_w,
                                           beta_w, g_w, gamma_w, s_w, whead, nullptr);

    // ---- erase/add + memory update ----
    gemm_wmma_kernel<<<4, 256, 0, stream>>>(h_t, H_, W_w, H_, b_w, nullptr,
                                            wa, 128, 2, H_, 0);
    erase_add_kernel<<<(B_ * 128) / 256, 256, 0, stream>>>(wa, e, a);
    mem_update_kernel<<<(B_ * N_ * 16) / 256, 256, 0, stream>>>(memory, whead, e, a, m_out);
}